// Model_8856222564745
// MI455X (gfx1250) — compile-verified
//
#include <hip/hip_runtime.h>
#include <hip/hip_bf16.h>

typedef __attribute__((ext_vector_type(2))) float v2f;
typedef __attribute__((ext_vector_type(8))) float v8f;

#define HDIM 100
#define HP   112   // feature stride padded to 7*16 so all WMMA tiles are full
#define EREL 100000
#define ELAB 200000

__device__ __forceinline__ v8f wmma4_f32(v2f a, v2f b, v8f c) {
  // V_WMMA_F32_16X16X4_F32: D = A(16x4) * B(4x16) + C(16x16), all fp32
  return __builtin_amdgcn_wmma_f32_16x16x4_f32(false, a, false, b, (short)0, c,
                                               false, false);
}

__device__ __forceinline__ float lrelu(float v) { return v > 0.f ? v : 0.01f * v; }

// ---------------------------------------------------------------- utilities
__global__ void fill0_k(float* __restrict__ p, int n) {
  int i = blockIdx.x * blockDim.x + threadIdx.x;
  if (i < n) p[i] = 0.f;
}

// copy [nrows x 100] -> padded [nrowsPad x HP], zero-filling pads
__global__ void pad_copy_k(float* __restrict__ dst, const float* __restrict__ src,
                           int nrows, int nrowsPad) {
  int i = blockIdx.x * blockDim.x + threadIdx.x;
  if (i >= nrowsPad * HP) return;
  int n = i / HP;
  int h = i - n * HP;
  dst[i] = (n < nrows && h < HDIM) ? src[n * HDIM + h] : 0.f;
}

// pack weights K-pair-interleaved: src [nmat][K][Nsrc] -> dst [nmat][K/2][N][2]
// dst[((mat*K + k)>>1 ... )] : flat = mat*K*N + (k>>1)*2*N + col*2 + (k&1)
// so for even ka, the pair (ka,col),(ka+1,col) sits at dst + mat*K*N + ka*N + col*2
__global__ void pack_w_k(float* __restrict__ dst, const float* __restrict__ src,
                         int nmat, int K, int N, int Nsrc) {
  long i = (long)blockIdx.x * blockDim.x + threadIdx.x;
  if (i >= (long)nmat * K * N) return;
  int col = (int)(i % N);
  long mk = i / N;
  int k = (int)(mk % K);
  int mat = (int)(mk / K);
  float v = (col < Nsrc) ? src[((long)mat * K + k) * Nsrc + col] : 0.f;
  dst[(long)mat * K * N + (long)(k >> 1) * 2 * N + col * 2 + (k & 1)] = v;
}

// pad biases: src [nmat][100] -> dst [nmat][HP]
__global__ void pad_b_k(float* __restrict__ dst, const float* __restrict__ src, int nmat) {
  int i = blockIdx.x * blockDim.x + threadIdx.x;
  if (i >= nmat * HP) return;
  int col = i % HP;
  int mat = i / HP;
  dst[i] = (col < HDIM) ? src[mat * HDIM + col] : 0.f;
}

__global__ void count_k(const int* __restrict__ dst, float* __restrict__ cnt, int E) {
  int e = blockIdx.x * blockDim.x + threadIdx.x;
  if (e < E) atomicAdd(&cnt[dst[e]], 1.0f);
}

// scatter-add messages: agg[dst[e], h..h+3] += x[src[e], h..h+3]
__global__ void scatter_k(const float* __restrict__ xs, const int* __restrict__ src,
                          const int* __restrict__ dst, float* __restrict__ agg, int E) {
  int i = blockIdx.x * blockDim.x + threadIdx.x;
  if (i >= E * 25) return;
  int e = i / 25;
  int h = (i - e * 25) * 4;
  const float4 v = *(const float4*)(xs + (long)src[e] * HP + h);
  float* dp = agg + (long)dst[e] * HP + h;
  atomicAdd(dp + 0, v.x);
  atomicAdd(dp + 1, v.y);
  atomicAdd(dp + 2, v.z);
  atomicAdd(dp + 3, v.w);
}

// x1 = disease_x @ W_dis + b_dis + disease_emb  (into padded layout)
__global__ void dis_init_k(const float* __restrict__ dx, const float* __restrict__ Wd,
                           const float* __restrict__ bd, const float* __restrict__ demb,
                           float* __restrict__ out, int nrows, int nrowsPad) {
  int i = blockIdx.x * blockDim.x + threadIdx.x;
  if (i >= nrowsPad * HP) return;
  int n = i / HP;
  int h = i - n * HP;
  float s = 0.f;
  if (n < nrows && h < HDIM) {
    s = bd[h] + demb[n * HDIM + h];
#pragma unroll
    for (int k = 0; k < 10; ++k) s = fmaf(dx[n * 10 + k], Wd[k * HDIM + h], s);
  }
  out[i] = s;
}

__global__ void leaky_k(float* __restrict__ dst, const float* __restrict__ src, int n) {
  int i = blockIdx.x * blockDim.x + threadIdx.x;
  if (i < n) dst[i] = lrelu(src[i]);
}

// ------------------------------------------------- per-relation SAGE GEMM
// out[64 x 16 slab] += (agg/cnt) @ Wl + xd @ Wr + bl
// one wave per block; 4 M-tiles share each B fragment (4-way WMMA ILP);
// B matrices pre-packed pair-interleaved so each fragment is one b64 load.
__global__ __launch_bounds__(32) void rel_gemm_k(
    const float* __restrict__ agg, const float* __restrict__ cnt,
    const float* __restrict__ xd, const float* __restrict__ Wlp,
    const float* __restrict__ Wrp, const float* __restrict__ blp,
    float* __restrict__ out) {
  int lane = threadIdx.x;
  int m0 = blockIdx.x * 64;
  int n0 = blockIdx.y * 16;
  int lrow = lane & 15;
  int col = n0 + lrow;
  int khalf = (lane >> 4) * 2;         // lanes 16-31 hold K+2,K+3
  int crow0 = m0 + (lane >> 4) * 8;    // C: VGPR v -> row crow0+v

  float inv[4];
#pragma unroll
  for (int t = 0; t < 4; ++t)
    inv[t] = 1.0f / fmaxf(cnt[m0 + t * 16 + lrow], 1.0f);

  // seed with current contents of out (accumulate across relations)
  v8f acc[4];
#pragma unroll
  for (int t = 0; t < 4; ++t)
#pragma unroll
    for (int v = 0; v < 8; ++v)
      acc[t][v] = out[(crow0 + t * 16 + v) * HP + col];

  // mean_neighbors @ Wl
#pragma unroll
  for (int k0 = 0; k0 < HDIM; k0 += 4) {
    int ka = k0 + khalf;
    v2f b = *(const v2f*)(Wlp + ka * HP + col * 2);   // pair-packed, 8B aligned
#pragma unroll
    for (int t = 0; t < 4; ++t) {
      v2f a = *(const v2f*)(agg + (m0 + t * 16 + lrow) * HP + ka);
      a = a * inv[t];
      acc[t] = wmma4_f32(a, b, acc[t]);
    }
  }
  // root @ Wr
#pragma unroll
  for (int k0 = 0; k0 < HDIM; k0 += 4) {
    int ka = k0 + khalf;
    v2f b = *(const v2f*)(Wrp + ka * HP + col * 2);
#pragma unroll
    for (int t = 0; t < 4; ++t) {
      v2f a = *(const v2f*)(xd + (m0 + t * 16 + lrow) * HP + ka);
      acc[t] = wmma4_f32(a, b, acc[t]);
    }
  }

  float bias = blp[col];
#pragma unroll
  for (int t = 0; t < 4; ++t)
#pragma unroll
    for (int v = 0; v < 8; ++v)
      out[(crow0 + t * 16 + v) * HP + col] = acc[t][v] + bias;
}

// ------------------------------------------------- fused link-pred MLP
#define FS  202   // even strides (8B-aligned pair loads), conflict-free mod 64
#define FS1 130
#define FS2 34
#define FS3 18

__global__ __launch_bounds__(256) void mlp_k(
    const float* __restrict__ x0, const float* __restrict__ x1,
    const int* __restrict__ idx_drug, const int* __restrict__ idx_dis,
    const float* __restrict__ W1p, const float* __restrict__ b1,
    const float* __restrict__ W2p, const float* __restrict__ b2,
    const float* __restrict__ W3p, const float* __restrict__ b3,
    const float* __restrict__ W4, const float* __restrict__ b4,
    float* __restrict__ out) {
  __shared__ float feat[16 * FS];
  __shared__ float h1s[16 * FS1];
  __shared__ float h2s[16 * FS2];
  __shared__ float h3s[16 * FS3];

  int tid = threadIdx.x;
  int base = blockIdx.x * 16;

  // gather concat([x0[drug], x1[dis]]) -> LDS, 16 edges x 200 feats
  for (int i = tid; i < 16 * 200; i += 256) {
    int e = i / 200;
    int k = i - e * 200;
    int ge = base + e;
    float v;
    if (k < HDIM) v = x0[(long)idx_drug[ge] * HP + k];
    else          v = x1[(long)idx_dis[ge] * HP + (k - HDIM)];
    feat[e * FS + k] = v;
  }
  __syncthreads();

  int wave = tid >> 5, lane = tid & 31;
  int mrow = lane & 15;
  int khalf = (lane >> 4) * 2;
  int crow = (lane >> 4) * 8;

  // layer 1: h1[16x128] = act(feat[16x200] @ W1 + b1); 8 waves, one n-tile each
  {
    int col = wave * 16 + mrow;
    v8f acc = {};
#pragma unroll
    for (int k0 = 0; k0 < 200; k0 += 4) {
      int ka = k0 + khalf;
      v2f a = *(const v2f*)(feat + mrow * FS + ka);
      v2f b = *(const v2f*)(W1p + ka * 128 + col * 2);
      acc = wmma4_f32(a, b, acc);
    }
    float bias = b1[col];
#pragma unroll
    for (int v = 0; v < 8; ++v)
      h1s[(crow + v) * FS1 + col] = lrelu(acc[v] + bias);
  }
  __syncthreads();

  // layer 2: h2[16x32] = act(h1 @ W2 + b2); waves 0-1
  if (wave < 2) {
    int col = wave * 16 + mrow;
    v8f acc = {};
#pragma unroll
    for (int k0 = 0; k0 < 128; k0 += 4) {
      int ka = k0 + khalf;
      v2f a = *(const v2f*)(h1s + mrow * FS1 + ka);
      v2f b = *(const v2f*)(W2p + ka * 32 + col * 2);
      acc = wmma4_f32(a, b, acc);
    }
    float bias = b2[col];
#pragma unroll
    for (int v = 0; v < 8; ++v)
      h2s[(crow + v) * FS2 + col] = lrelu(acc[v] + bias);
  }
  __syncthreads();

  // layer 3: h3[16x16] = act(h2 @ W3 + b3); wave 0
  if (wave == 0) {
    int col = mrow;
    v8f acc = {};
#pragma unroll
    for (int k0 = 0; k0 < 32; k0 += 4) {
      int ka = k0 + khalf;
      v2f a = *(const v2f*)(h2s + mrow * FS2 + ka);
      v2f b = *(const v2f*)(W3p + ka * 16 + col * 2);
      acc = wmma4_f32(a, b, acc);
    }
    float bias = b3[col];
#pragma unroll
    for (int v = 0; v < 8; ++v)
      h3s[(crow + v) * FS3 + col] = lrelu(acc[v] + bias);
  }
  __syncthreads();

  // final: out = h3 @ W4 + b4
  if (tid < 16) {
    float s = b4[0];
#pragma unroll
    for (int k = 0; k < 16; ++k) s = fmaf(h3s[tid * FS3 + k], W4[k], s);
    out[base + tid] = s;
  }
}

// ---------------------------------------------------------------- driver
extern "C" void kernel_launch(void* const* d_in, const int* in_sizes, int n_in,
                              void* d_out, int out_size, void* d_ws, size_t ws_size,
                              hipStream_t stream) {
  const float* drug_emb    = (const float*)d_in[0];
  const float* disease_emb = (const float*)d_in[1];
  const float* disease_x   = (const float*)d_in[2];
  const float* W_dis       = (const float*)d_in[3];
  const float* b_dis       = (const float*)d_in[4];
  const float* attr_embs   = (const float*)d_in[5];
  const float* Wl          = (const float*)d_in[6];
  const float* bl          = (const float*)d_in[7];
  const float* Wr          = (const float*)d_in[8];
  const float* W1 = (const float*)d_in[9];
  const float* b1 = (const float*)d_in[10];
  const float* W2 = (const float*)d_in[11];
  const float* b2 = (const float*)d_in[12];
  const float* W3 = (const float*)d_in[13];
  const float* b3 = (const float*)d_in[14];
  const float* W4 = (const float*)d_in[15];
  const float* b4 = (const float*)d_in[16];
  const int* edge_index = (const int*)d_in[17];
  const int* eli        = (const int*)d_in[18];
  float* out = (float*)d_out;

  static const int NT[10] = {50000, 10000, 2000, 2000, 2000, 2000, 2000, 2000, 2000, 2000};
  static const int RP[10] = {50048, 10048, 2048, 2048, 2048, 2048, 2048, 2048, 2048, 2048};
  static const int RS[18] = {0, 1, 0, 2, 0, 3, 0, 4, 0, 5, 0, 6, 0, 7, 0, 8, 0, 9};
  static const int RD[18] = {1, 0, 2, 0, 3, 0, 4, 0, 5, 0, 6, 0, 7, 0, 8, 0, 9, 0};

  long toff[10];
  long tot = 0;
  for (int t = 0; t < 10; ++t) { toff[t] = tot; tot += (long)RP[t] * HP; }  // 8.57e6 floats

  float* xA = (float*)d_ws;                    // current activations (padded)
  float* xB = xA + tot;                        // layer accumulator (padded)
  float* agg = xB + tot;                       // scatter buffer (max 50048 x HP)
  float* counts = agg + (long)50048 * HP;      // padded per-relation in-degree
  long coff[18];
  long ctot = 0;
  for (int r = 0; r < 18; ++r) { coff[r] = ctot; ctot += RP[RD[r]]; }
  float* Wlp = counts + ctot;                  // 72 x 100 x HP (pair-packed)
  float* Wrp = Wlp + (long)72 * HDIM * HP;
  float* blp = Wrp + (long)72 * HDIM * HP;     // 72 x HP
  float* W1p = blp + 72 * HP;                  // 200x128 pair-packed
  float* W2p = W1p + 200 * 128;                // 128x32 pair-packed
  float* W3p = W2p + 128 * 32;                 // 32x16 pair-packed

  auto nb = [](long n, int bs) { return (unsigned)((n + bs - 1) / bs); };

  // pack weights / pad biases once per call
  pack_w_k<<<nb((long)72 * HDIM * HP, 256), 256, 0, stream>>>(Wlp, Wl, 72, HDIM, HP, HDIM);
  pack_w_k<<<nb((long)72 * HDIM * HP, 256), 256, 0, stream>>>(Wrp, Wr, 72, HDIM, HP, HDIM);
  pad_b_k<<<nb(72 * HP, 256), 256, 0, stream>>>(blp, bl, 72);
  pack_w_k<<<nb(200 * 128, 256), 256, 0, stream>>>(W1p, W1, 1, 200, 128, 128);
  pack_w_k<<<nb(128 * 32, 256), 256, 0, stream>>>(W2p, W2, 1, 128, 32, 32);
  pack_w_k<<<nb(32 * 16, 256), 256, 0, stream>>>(W3p, W3, 1, 32, 16, 16);

  // in-degree counts per relation (padded, zero-filled tails)
  fill0_k<<<nb(ctot, 256), 256, 0, stream>>>(counts, (int)ctot);
  for (int r = 0; r < 18; ++r) {
    const int* dst = edge_index + ((long)r * 2 + 1) * EREL;
    count_k<<<nb(EREL, 256), 256, 0, stream>>>(dst, counts + coff[r], EREL);
  }

  // initial node features (padded)
  pad_copy_k<<<nb((long)RP[0] * HP, 256), 256, 0, stream>>>(xA + toff[0], drug_emb,
                                                            NT[0], RP[0]);
  dis_init_k<<<nb((long)RP[1] * HP, 256), 256, 0, stream>>>(disease_x, W_dis, b_dis,
                                                            disease_emb, xA + toff[1],
                                                            NT[1], RP[1]);
  for (int a = 0; a < 8; ++a)
    pad_copy_k<<<nb((long)RP[2 + a] * HP, 256), 256, 0, stream>>>(
        xA + toff[2 + a], attr_embs + (long)a * 2000 * HDIM, NT[2 + a], RP[2 + a]);

  for (int layer = 0; layer < 4; ++layer) {
    fill0_k<<<nb(tot, 256), 256, 0, stream>>>(xB, (int)tot);
    for (int r = 0; r < 18; ++r) {
      int s = RS[r], d = RD[r];
      const int* srcI = edge_index + ((long)r * 2 + 0) * EREL;
      const int* dstI = edge_index + ((long)r * 2 + 1) * EREL;
      fill0_k<<<nb((long)RP[d] * HP, 256), 256, 0, stream>>>(agg, RP[d] * HP);
      scatter_k<<<nb((long)EREL * 25, 256), 256, 0, stream>>>(xA + toff[s], srcI, dstI,
                                                              agg, EREL);
      dim3 g((unsigned)(RP[d] / 64), (unsigned)(HP / 16));
      rel_gemm_k<<<g, 32, 0, stream>>>(agg, counts + coff[r], xA + toff[d],
                                       Wlp + ((long)layer * 18 + r) * HDIM * HP,
                                       Wrp + ((long)layer * 18 + r) * HDIM * HP,
                                       blp + ((long)layer * 18 + r) * HP,
                                       xB + toff[d]);
    }
    leaky_k<<<nb(tot, 256), 256, 0, stream>>>(xA, xB, (int)tot);
  }

  mlp_k<<<ELAB / 16, 256, 0, stream>>>(xA + toff[0], xA + toff[1], eli, eli + ELAB,
                                       W1p, b1, W2p, b2, W3p, b3, W4, b4, out);
}